// Model_40630390620466
// MI455X (gfx1250) — compile-verified
//
#include <hip/hip_runtime.h>
#include <hip/hip_bf16.h>

typedef __attribute__((ext_vector_type(2))) float v2f;
typedef __attribute__((ext_vector_type(8))) float v8f;

namespace {
constexpr int Bn = 8;        // batch
constexpr int Sn = 4096;     // sequence
constexpr int Hn = 16;       // heads
constexpr int Pn = 64;       // head dim
constexpr int Nn = 64;       // state dim
constexpr int Ln = 64;       // chunk length
constexpr int Cn = Sn / Ln;  // chunks = 64
constexpr int PAD = 68;      // LDS row pitch (floats): 68%64=4 -> conflict-free strided rows
}

// ---------------------------------------------------------------------------
// Kernel 1: per-(b,c,h) local chunk states:
//   states[p][n] = sum_l X[l][p] * exp(Acum[63]-Acum[l]) * B[l][n]
// and store Asum (=Acum[63]) for the inter-chunk scan.
// ---------------------------------------------------------------------------
__global__ __launch_bounds__(128) void ssd_chunk_states(
    const float* __restrict__ X, const float* __restrict__ B,
    const float* __restrict__ A, float* __restrict__ states,
    float* __restrict__ asum)
{
  const int ci = blockIdx.x, hi = blockIdx.y, bi = blockIdx.z;
  __shared__ float Xs[Ln][PAD];
  __shared__ float Bs[Ln][PAD];
  __shared__ float Acum[Ln];
  __shared__ float dec[Ln];

  const int tid = threadIdx.x;
  const size_t gbase = ((size_t)(bi * Sn + ci * Ln) * Hn + hi) * Pn;
  const size_t gstride = (size_t)Hn * Pn;

  // stage X, B tiles (64x64) into LDS, float4 coalesced
  for (int it = 0; it < 8; ++it) {
    int idx = tid + 128 * it;
    int row = idx >> 4, c4 = (idx & 15) * 4;
    *(float4*)&Xs[row][c4] = *(const float4*)&X[gbase + row * gstride + c4];
    *(float4*)&Bs[row][c4] = *(const float4*)&B[gbase + row * gstride + c4];
  }
  if (tid < Ln) Acum[tid] = A[(size_t)(bi * Sn + ci * Ln + tid) * Hn + hi];
  __syncthreads();
  if (tid == 0) {  // tiny serial cumsum (64 adds)
    float s = 0.f;
    for (int i = 0; i < Ln; ++i) { s += Acum[i]; Acum[i] = s; }
  }
  __syncthreads();
  if (tid < Ln) dec[tid] = __expf(Acum[Ln - 1] - Acum[tid]);
  __syncthreads();

  const int lane = tid & 31, wv = tid >> 5;
  const int r = lane & 15, hf = lane >> 4;
  const int p0 = wv * 16;  // each wave owns a 16-row strip of the p dimension

  float* sb = states + ((size_t)(bi * Cn + ci) * Hn + hi) * (Pn * Nn);
  for (int n0 = 0; n0 < Nn; n0 += 16) {
    v8f acc = {};
    for (int k0 = 0; k0 < Ln; k0 += 4) {
      int k = k0 + 2 * hf;
      v2f a, bf;
      a.x = Xs[k][p0 + r] * dec[k];         // A-op[m=p][k=l] = X^T scaled
      a.y = Xs[k + 1][p0 + r] * dec[k + 1];
      bf.x = Bs[k][n0 + r];                 // B-op[k=l][col=n]
      bf.y = Bs[k + 1][n0 + r];
      acc = __builtin_amdgcn_wmma_f32_16x16x4_f32(false, a, false, bf,
                                                  (short)0, acc, false, false);
    }
#pragma unroll
    for (int v = 0; v < 8; ++v) {
      int prow = p0 + v + 8 * hf;
      sb[prow * Nn + n0 + r] = acc[v];
    }
  }
  if (tid == 0) asum[((size_t)bi * Hn + hi) * Cn + ci] = Acum[Ln - 1];
}

// ---------------------------------------------------------------------------
// Kernel 2: sequential inter-chunk scan per (b,h), in place over `states`:
//   P[0] = initial;  P[z] = exp(Asum[z-1]) * P[z-1] + local[z-1]
// Each thread keeps its 16 state elements (and previous local) in registers.
// ---------------------------------------------------------------------------
__global__ __launch_bounds__(256) void ssd_state_scan(
    float* __restrict__ states, const float* __restrict__ init,
    const float* __restrict__ asum)
{
  const int hi = blockIdx.x, bi = blockIdx.y;
  __shared__ float As[Cn];
  const int tid = threadIdx.x;
  if (tid < Cn) As[tid] = asum[((size_t)bi * Hn + hi) * Cn + tid];
  __syncthreads();

  const float* ib = init + ((size_t)bi * Hn + hi) * (Pn * Nn);
  float Pprev[16], locPrev[16], Pz[16], loc[16];

  for (int z = 0; z < Cn; ++z) {
    float* sb = states + ((size_t)(bi * Cn + z) * Hn + hi) * (Pn * Nn);
#pragma unroll
    for (int j = 0; j < 16; ++j) loc[j] = sb[tid + 256 * j];
    if (z == 0) {
#pragma unroll
      for (int j = 0; j < 16; ++j) Pz[j] = ib[tid + 256 * j];
    } else {
      float ez = __expf(As[z - 1]);
#pragma unroll
      for (int j = 0; j < 16; ++j) Pz[j] = ez * Pprev[j] + locPrev[j];
    }
#pragma unroll
    for (int j = 0; j < 16; ++j) sb[tid + 256 * j] = Pz[j];
#pragma unroll
    for (int j = 0; j < 16; ++j) { Pprev[j] = Pz[j]; locPrev[j] = loc[j]; }
  }
}

// ---------------------------------------------------------------------------
// Kernel 3: per-(b,c,h) output:
//   G[l][s] = (l>=s) ? exp(Acum[l]-Acum[s]) * sum_n C[l][n]B[s][n] : 0
//   Y[l][p] = sum_s G[l][s] X[s][p]  +  exp(Acum[l]) * sum_n C[l][n] P[p][n]
// Only 3 live 64x68 LDS buffers via phase reuse (52.7 KB).
// ---------------------------------------------------------------------------
__global__ __launch_bounds__(128) void ssd_output(
    const float* __restrict__ X, const float* __restrict__ B,
    const float* __restrict__ C, const float* __restrict__ A,
    const float* __restrict__ states, float* __restrict__ Y)
{
  const int ci = blockIdx.x, hi = blockIdx.y, bi = blockIdx.z;
  __shared__ float M0[Ln][PAD];  // C -> (C scaled by eA) -> X
  __shared__ float M1[Ln][PAD];  // B -> P
  __shared__ float M2[Ln][PAD];  // G
  __shared__ float Acum[Ln];
  __shared__ float eA[Ln];

  const int tid = threadIdx.x;
  const size_t gbase = ((size_t)(bi * Sn + ci * Ln) * Hn + hi) * Pn;
  const size_t gstride = (size_t)Hn * Pn;

  for (int it = 0; it < 8; ++it) {
    int idx = tid + 128 * it;
    int row = idx >> 4, c4 = (idx & 15) * 4;
    *(float4*)&M0[row][c4] = *(const float4*)&C[gbase + row * gstride + c4];
    *(float4*)&M1[row][c4] = *(const float4*)&B[gbase + row * gstride + c4];
  }
  if (tid < Ln) Acum[tid] = A[(size_t)(bi * Sn + ci * Ln + tid) * Hn + hi];
  __syncthreads();
  if (tid == 0) {
    float s = 0.f;
    for (int i = 0; i < Ln; ++i) { s += Acum[i]; Acum[i] = s; }
  }
  __syncthreads();
  if (tid < Ln) eA[tid] = __expf(Acum[tid]);
  __syncthreads();

  const int lane = tid & 31, wv = tid >> 5;
  const int r = lane & 15, hf = lane >> 4;
  const int l0 = wv * 16;  // wave owns output rows l0..l0+15

  // --- Phase 1: G = mask(C * B^T) into M2 (skip fully-masked upper tiles) ---
  for (int s0 = 0; s0 < Ln; s0 += 16) {
    if (s0 > l0) {
#pragma unroll
      for (int v = 0; v < 8; ++v) M2[l0 + v + 8 * hf][s0 + r] = 0.0f;
      continue;
    }
    v8f acc = {};
    for (int k0 = 0; k0 < Nn; k0 += 4) {  // contract over n
      int k = k0 + 2 * hf;
      v2f a, bf;
      a.x = M0[l0 + r][k];      a.y = M0[l0 + r][k + 1];   // C rows
      bf.x = M1[s0 + r][k];     bf.y = M1[s0 + r][k + 1];  // B rows (B^T cols)
      acc = __builtin_amdgcn_wmma_f32_16x16x4_f32(false, a, false, bf,
                                                  (short)0, acc, false, false);
    }
#pragma unroll
    for (int v = 0; v < 8; ++v) {
      int row = l0 + v + 8 * hf, col = s0 + r;
      M2[row][col] = (row >= col) ? acc[v] * __expf(Acum[row] - Acum[col]) : 0.0f;
    }
  }
  __syncthreads();

  // --- reload: M1 <- P (scanned state), M0 <- C * eA[row] ---
  {
    const float* pb = states + ((size_t)(bi * Cn + ci) * Hn + hi) * (Pn * Nn);
    for (int it = 0; it < 8; ++it) {
      int idx = tid + 128 * it;
      int row = idx >> 4, c4 = (idx & 15) * 4;
      *(float4*)&M1[row][c4] = *(const float4*)&pb[row * Nn + c4];
    }
    for (int it = 0; it < 32; ++it) {
      int idx = tid + 128 * it;
      int row = idx >> 6, col = idx & 63;
      M0[row][col] *= eA[row];
    }
  }
  __syncthreads();

  // --- Phase 2a: Y_off = (C*eA) @ P^T, accumulate in registers ---
  v8f acc2[4];
#pragma unroll
  for (int pi = 0; pi < 4; ++pi) {
    int p0 = pi * 16;
    v8f acc = {};
    for (int k0 = 0; k0 < Nn; k0 += 4) {  // contract over n
      int k = k0 + 2 * hf;
      v2f a, bf;
      a.x = M0[l0 + r][k];   a.y = M0[l0 + r][k + 1];   // scaled C rows
      bf.x = M1[p0 + r][k];  bf.y = M1[p0 + r][k + 1];  // P[p][n] rows
      acc = __builtin_amdgcn_wmma_f32_16x16x4_f32(false, a, false, bf,
                                                  (short)0, acc, false, false);
    }
    acc2[pi] = acc;
  }
  __syncthreads();

  // --- reload: M0 <- X ---
  for (int it = 0; it < 8; ++it) {
    int idx = tid + 128 * it;
    int row = idx >> 4, c4 = (idx & 15) * 4;
    *(float4*)&M0[row][c4] = *(const float4*)&X[gbase + row * gstride + c4];
  }
  __syncthreads();

  // --- Phase 2b: Y += G @ X (K truncated at the diagonal block), write out ---
#pragma unroll
  for (int pi = 0; pi < 4; ++pi) {
    int p0 = pi * 16;
    v8f acc = acc2[pi];
    for (int k0 = 0; k0 < l0 + 16; k0 += 4) {  // contract over s (causal)
      int k = k0 + 2 * hf;
      v2f a, bf;
      a.x = M2[l0 + r][k];   a.y = M2[l0 + r][k + 1];   // G rows
      bf.x = M0[k][p0 + r];  bf.y = M0[k + 1][p0 + r];  // X[s][p]
      acc = __builtin_amdgcn_wmma_f32_16x16x4_f32(false, a, false, bf,
                                                  (short)0, acc, false, false);
    }
#pragma unroll
    for (int v = 0; v < 8; ++v) {
      int row = l0 + v + 8 * hf;
      Y[gbase + row * gstride + p0 + r] = acc[v];
    }
  }
}

// ---------------------------------------------------------------------------
extern "C" void kernel_launch(void* const* d_in, const int* in_sizes, int n_in,
                              void* d_out, int out_size, void* d_ws, size_t ws_size,
                              hipStream_t stream) {
  (void)in_sizes; (void)n_in; (void)out_size; (void)ws_size;
  const float* X    = (const float*)d_in[0];
  const float* init = (const float*)d_in[1];
  const float* A    = (const float*)d_in[2];
  const float* Bm   = (const float*)d_in[3];
  const float* Cm   = (const float*)d_in[4];
  // d_in[5] = chunk_size (==64, hardcoded)
  float* Y = (float*)d_out;

  float* states = (float*)d_ws;                          // b*c*h*p*n floats (134 MB)
  float* asum   = states + (size_t)Bn * Cn * Hn * Pn * Nn;  // b*h*c floats

  dim3 grd(Cn, Hn, Bn);
  ssd_chunk_states<<<grd, 128, 0, stream>>>(X, Bm, A, states, asum);
  ssd_state_scan<<<dim3(Hn, Bn), 256, 0, stream>>>(states, init, asum);
  ssd_output<<<grd, 128, 0, stream>>>(X, Bm, Cm, A, states, Y);
}